// TargetEdgeInitializer_25537875542501
// MI455X (gfx1250) — compile-verified
//
#include <hip/hip_runtime.h>
#include <math.h>

#define NN 20000
#define NE 320000
#define NHEAD 4

typedef float v2f __attribute__((ext_vector_type(2)));
typedef float v8f __attribute__((ext_vector_type(8)));

__device__ __forceinline__ int enc_f32(float f) {
    int b = __float_as_int(f);
    return b >= 0 ? b : (b ^ 0x7fffffff);
}
__device__ __forceinline__ float dec_f32(int b) {
    return __int_as_float(b >= 0 ? b : (b ^ 0x7fffffff));
}

// ---------------- fill kernels ----------------
__global__ void fill_f32(float* p, int n, float v) {
    int t = blockIdx.x * blockDim.x + threadIdx.x;
    if (t < n) p[t] = v;
}
__global__ void fill_i32(int* p, int n, int v) {
    int t = blockIdx.x * blockDim.x + threadIdx.x;
    if (t < n) p[t] = v;
}

// ---------------- WMMA GEMM: C[rows,dout] = A[rows,K] @ B[K,dout] + bias ----
// one wave -> 32 (M) x 64 (N) register tile, fp32 WMMA 16x16x4 (fp32-exact)
// per k-step: 2 x b64 (A) + 8 x b32 (B) loads feeding 8 WMMAs
__global__ __launch_bounds__(256) void gemm_wmma(
    const float* __restrict__ A, const float* __restrict__ B,
    const float* __restrict__ bias, float* __restrict__ C,
    int rows, int K, int dout)
{
    int lane = threadIdx.x & 31;
    int wave = threadIdx.x >> 5;
    int tilesN = dout >> 6;                 // strips of 64 cols
    int w = blockIdx.x * 8 + wave;
    int tm = w / tilesN, tn = w % tilesN;   // tm over rows/32
    if (tm * 32 >= rows) return;

    int kb  = (lane >> 4) << 1;             // 0 or 2
    int n0  = tn * 64 + (lane & 15);
    int mr0 = tm * 32 + (lane & 15);
    int mr1 = mr0 + 16;

    v8f c00 = {}, c01 = {}, c02 = {}, c03 = {};
    v8f c10 = {}, c11 = {}, c12 = {}, c13 = {};

    const float* ap0 = A + (size_t)mr0 * K + kb;
    const float* ap1 = A + (size_t)mr1 * K + kb;
    const float* bp  = B + (size_t)kb * dout + n0;
    const size_t bstep = (size_t)dout * 4;

    for (int k = 0; k < K; k += 4) {
        // gather all fragments first -> one load clause, one wait, 8 WMMAs
        v2f a0; a0.x = ap0[0];  a0.y = ap0[1];
        v2f a1; a1.x = ap1[0];  a1.y = ap1[1];
        v2f b0; b0.x = bp[0];   b0.y = bp[dout];
        v2f b1; b1.x = bp[16];  b1.y = bp[dout + 16];
        v2f b2; b2.x = bp[32];  b2.y = bp[dout + 32];
        v2f b3; b3.x = bp[48];  b3.y = bp[dout + 48];
        c00 = __builtin_amdgcn_wmma_f32_16x16x4_f32(false, a0, false, b0, (short)0, c00, false, false);
        c01 = __builtin_amdgcn_wmma_f32_16x16x4_f32(false, a0, false, b1, (short)0, c01, false, false);
        c02 = __builtin_amdgcn_wmma_f32_16x16x4_f32(false, a0, false, b2, (short)0, c02, false, false);
        c03 = __builtin_amdgcn_wmma_f32_16x16x4_f32(false, a0, false, b3, (short)0, c03, false, false);
        c10 = __builtin_amdgcn_wmma_f32_16x16x4_f32(false, a1, false, b0, (short)0, c10, false, false);
        c11 = __builtin_amdgcn_wmma_f32_16x16x4_f32(false, a1, false, b1, (short)0, c11, false, false);
        c12 = __builtin_amdgcn_wmma_f32_16x16x4_f32(false, a1, false, b2, (short)0, c12, false, false);
        c13 = __builtin_amdgcn_wmma_f32_16x16x4_f32(false, a1, false, b3, (short)0, c13, false, false);
        ap0 += 4; ap1 += 4; bp += bstep;
    }

    float bv0 = bias ? bias[n0]      : 0.f;
    float bv1 = bias ? bias[n0 + 16] : 0.f;
    float bv2 = bias ? bias[n0 + 32] : 0.f;
    float bv3 = bias ? bias[n0 + 48] : 0.f;
    int r0 = tm * 32 + ((lane >> 4) << 3);  // +0 or +8 within first 16-row tile
#pragma unroll
    for (int r = 0; r < 8; r++) {
        size_t ra = (size_t)(r0 + r) * dout;
        size_t rb = (size_t)(r0 + 16 + r) * dout;
        C[ra + n0]      = c00[r] + bv0;
        C[ra + n0 + 16] = c01[r] + bv1;
        C[ra + n0 + 32] = c02[r] + bv2;
        C[ra + n0 + 48] = c03[r] + bv3;
        C[rb + n0]      = c10[r] + bv0;
        C[rb + n0 + 16] = c11[r] + bv1;
        C[rb + n0 + 32] = c12[r] + bv2;
        C[rb + n0 + 48] = c13[r] + bv3;
    }
}

// ---------------- WMMA SYRK: G += X[k0:k0+klen,:]^T @ X[k0:k0+klen,:] ------
// one wave -> 32 x 32 tile (4 accumulators), split-K via blockIdx.y
__global__ __launch_bounds__(256) void syrk_wmma(
    const float* __restrict__ X, float* __restrict__ G, int dcols, int klen)
{
    int lane = threadIdx.x & 31;
    int wave = threadIdx.x >> 5;
    int tiles = dcols >> 5;                 // 16 strips of 32
    int w = blockIdx.x * 8 + wave;
    int tm = w / tiles, tn = w % tiles;
    if (tm >= tiles) return;
    int k0 = blockIdx.y * klen;

    int gm0 = tm * 32 + (lane & 15), gm1 = gm0 + 16;
    int gn0 = tn * 32 + (lane & 15), gn1 = gn0 + 16;
    int kb = (lane >> 4) << 1;

    v8f c00 = {}, c01 = {}, c10 = {}, c11 = {};
    const float* rp = X + (size_t)(k0 + kb) * dcols;
    const size_t rstep = (size_t)dcols * 4;
    for (int k = 0; k < klen; k += 4) {
        const float* r1 = rp + dcols;
        v2f a0; a0.x = rp[gm0]; a0.y = r1[gm0];
        v2f a1; a1.x = rp[gm1]; a1.y = r1[gm1];
        v2f b0; b0.x = rp[gn0]; b0.y = r1[gn0];
        v2f b1; b1.x = rp[gn1]; b1.y = r1[gn1];
        c00 = __builtin_amdgcn_wmma_f32_16x16x4_f32(false, a0, false, b0, (short)0, c00, false, false);
        c01 = __builtin_amdgcn_wmma_f32_16x16x4_f32(false, a0, false, b1, (short)0, c01, false, false);
        c10 = __builtin_amdgcn_wmma_f32_16x16x4_f32(false, a1, false, b0, (short)0, c10, false, false);
        c11 = __builtin_amdgcn_wmma_f32_16x16x4_f32(false, a1, false, b1, (short)0, c11, false, false);
        rp += rstep;
    }
    int rr0 = tm * 32 + ((lane >> 4) << 3);
#pragma unroll
    for (int r = 0; r < 8; r++) {
        size_t ra = (size_t)(rr0 + r) * dcols;
        size_t rb = (size_t)(rr0 + 16 + r) * dcols;
        atomicAdd(&G[ra + gn0], c00[r]);
        atomicAdd(&G[ra + gn1], c01[r]);
        atomicAdd(&G[rb + gn0], c10[r]);
        atomicAdd(&G[rb + gn1], c11[r]);
    }
}

// ---------------- attention edge kernels ----------------
__global__ void edge_alpha(
    const float* __restrict__ Q, const float* __restrict__ Kt,
    const float* __restrict__ We, const float* __restrict__ ea,
    const int* __restrict__ src, const int* __restrict__ dst,
    float* __restrict__ alpha, int* __restrict__ amax, int E, int dout, int dh)
{
    int t = blockIdx.x * blockDim.x + threadIdx.x;
    if (t >= E * NHEAD) return;
    int e = t >> 2, h = t & 3;
    int j = src[e], i = dst[e];
    float w = ea[e];
    const float4* q4 = (const float4*)(Q  + (size_t)i * dout + h * dh);
    const float4* k4 = (const float4*)(Kt + (size_t)j * dout + h * dh);
    const float4* w4 = (const float4*)(We + h * dh);
    float s = 0.f;
    int n4 = dh >> 2;
#pragma unroll 4
    for (int c = 0; c < n4; c++) {
        float4 qv = q4[c], kv = k4[c], wv = w4[c];
        s += qv.x * (kv.x + w * wv.x) + qv.y * (kv.y + w * wv.y)
           + qv.z * (kv.z + w * wv.z) + qv.w * (kv.w + w * wv.w);
    }
    s *= rsqrtf((float)dh);
    alpha[t] = s;
    atomicMax(&amax[i * NHEAD + h], enc_f32(s));
}

__global__ void edge_accum(
    const float* __restrict__ V, const float* __restrict__ We,
    const float* __restrict__ ea, const int* __restrict__ src,
    const int* __restrict__ dst, const float* __restrict__ alpha,
    const int* __restrict__ amax, float* __restrict__ den,
    float* __restrict__ ACC, int E, int dout, int dh)
{
    int t = blockIdx.x * blockDim.x + threadIdx.x;
    if (t >= E * NHEAD) return;
    int e = t >> 2, h = t & 3;
    int j = src[e], i = dst[e];
    float w = ea[e];
    float m = dec_f32(amax[i * NHEAD + h]);
    float ex = __expf(alpha[t] - m);
    atomicAdd(&den[i * NHEAD + h], ex);
    const float4* v4 = (const float4*)(V + (size_t)j * dout + h * dh);
    const float4* w4 = (const float4*)(We + h * dh);
    float* acc = ACC + (size_t)i * dout + h * dh;
    int n4 = dh >> 2;
#pragma unroll 4
    for (int c = 0; c < n4; c++) {
        float4 vv = v4[c], wv = w4[c];
        atomicAdd(&acc[4 * c + 0], (vv.x + w * wv.x) * ex);
        atomicAdd(&acc[4 * c + 1], (vv.y + w * wv.y) * ex);
        atomicAdd(&acc[4 * c + 2], (vv.z + w * wv.z) * ex);
        atomicAdd(&acc[4 * c + 3], (vv.w + w * wv.w) * ex);
    }
}

// XOUT already holds skip; add msg/den (+x_res for last layer)
__global__ void attn_finalize(
    float* __restrict__ XOUT, const float* __restrict__ ACC,
    const float* __restrict__ den, const float* __restrict__ XR,
    int rows, int dout, int dh)
{
    int t = blockIdx.x * blockDim.x + threadIdx.x;
    if (t >= rows * dout) return;
    int i = t / dout, c = t % dout;
    int h = c / dh;
    float d = den[i * NHEAD + h];
    float msg = d > 0.f ? ACC[t] / d : 0.f;
    float v = XOUT[t] + msg;
    if (XR) v += XR[t];
    XOUT[t] = v;
}

// ---------------- GraphNorm ----------------
__global__ void gn_stats(const float* __restrict__ X, float* colsum, float* colsq,
                         int rows, int dout, int rchunk)
{
    int c = blockIdx.x * blockDim.x + threadIdx.x;
    if (c >= dout) return;
    int rb = blockIdx.y * rchunk;
    int re = rb + rchunk; if (re > rows) re = rows;
    float s = 0.f, q = 0.f;
    for (int r = rb; r < re; r++) {
        float v = X[(size_t)r * dout + c];
        s += v; q += v * v;
    }
    atomicAdd(&colsum[c], s);
    atomicAdd(&colsq[c], q);
}

__global__ void gn_final(const float* colsum, const float* colsq,
                         const float* __restrict__ ms, float* mu, float* istd,
                         int rows, int dout)
{
    int c = blockIdx.x * blockDim.x + threadIdx.x;
    if (c >= dout) return;
    float m   = colsum[c] / (float)rows;
    float ex2 = colsq[c] / (float)rows;
    float s   = ms[c];
    float var = ex2 - 2.f * s * m * m + s * s * m * m;  // E[(x - s*mu)^2]
    mu[c] = m;
    istd[c] = rsqrtf(var + 1e-5f);
}

__global__ void gn_apply(const float* __restrict__ X, float* __restrict__ Y,
                         const float* mu, const float* istd, const float* ms,
                         const float* wgt, const float* bia, int rows, int dout)
{
    int t = blockIdx.x * blockDim.x + threadIdx.x;
    if (t >= rows * dout) return;
    int c = t % dout;
    float v = (X[t] - ms[c] * mu[c]) * istd[c] * wgt[c] + bia[c];
    Y[t] = v > 0.f ? v : 0.f;
}

// ---------------- GCN ----------------
__global__ void gcn_deg(const float* __restrict__ ea, const int* __restrict__ dst,
                        float* deg, int E)
{
    int t = blockIdx.x * blockDim.x + threadIdx.x;
    if (t < E) atomicAdd(&deg[dst[t]], ea[t]);
}
__global__ void gcn_dinv(const float* deg, float* dinv, int n)
{
    int t = blockIdx.x * blockDim.x + threadIdx.x;
    if (t < n) dinv[t] = deg[t] > 0.f ? rsqrtf(deg[t]) : 0.f;
}
__global__ void gcn_self(const float* __restrict__ h, const float* __restrict__ dinv,
                         const float* __restrict__ bg, float* __restrict__ xt,
                         int rows, int dout)
{
    int t = blockIdx.x * blockDim.x + threadIdx.x;
    if (t >= rows * dout) return;
    int i = t / dout, c = t % dout;
    float di = dinv[i];
    xt[t] = 2.f * di * di * h[t] + bg[c];
}
__global__ void gcn_scatter(const float* __restrict__ h, const float* __restrict__ dinv,
                            const float* __restrict__ ea, const int* __restrict__ src,
                            const int* __restrict__ dst, float* __restrict__ xt,
                            int E, int dout)
{
    const int CW = 16;
    int chunks = dout / CW;                 // 32
    int t = blockIdx.x * blockDim.x + threadIdx.x;
    if (t >= E * chunks) return;
    int e = t / chunks, ch = t % chunks;
    int j = src[e], i = dst[e];
    float nrm = dinv[j] * ea[e] * dinv[i];
    const float4* hs = (const float4*)(h + (size_t)j * dout + ch * CW);
    float* xo = xt + (size_t)i * dout + ch * CW;
#pragma unroll
    for (int c = 0; c < CW / 4; c++) {
        float4 hv = hs[c];
        atomicAdd(&xo[4 * c + 0], nrm * hv.x);
        atomicAdd(&xo[4 * c + 1], nrm * hv.y);
        atomicAdd(&xo[4 * c + 2], nrm * hv.z);
        atomicAdd(&xo[4 * c + 3], nrm * hv.w);
    }
}

// ---------------- min/max + triu ----------------
__global__ void minmax_init(int* mm) { mm[0] = 0x7fffffff; mm[1] = 0x80000000; }
__global__ void minmax_red(const float* __restrict__ g, int n, int* mm)
{
    int t = blockIdx.x * blockDim.x + threadIdx.x;
    int stride = gridDim.x * blockDim.x;
    int lmin = 0x7fffffff, lmax = 0x80000000;
    for (int i = t; i < n; i += stride) {
        int e = enc_f32(g[i]);
        lmin = e < lmin ? e : lmin;
        lmax = e > lmax ? e : lmax;
    }
    atomicMin(&mm[0], lmin);
    atomicMax(&mm[1], lmax);
}
__global__ void triu_out(const float* __restrict__ g, const int* __restrict__ mm,
                         float* __restrict__ out, int n)
{
    int t = blockIdx.x * blockDim.x + threadIdx.x;
    if (t >= n * n) return;
    int i = t / n, j = t % n;
    if (j <= i) return;
    float mn = dec_f32(mm[0]), mx = dec_f32(mm[1]);
    size_t off = (size_t)i * (2 * n - 1 - i) / 2;
    out[off + (j - i - 1)] = (g[t] - mn) / (mx - mn + 1e-8f);
}

// ---------------- host ----------------
extern "C" void kernel_launch(void* const* d_in, const int* in_sizes, int n_in,
                              void* d_out, int out_size, void* d_ws, size_t ws_size,
                              hipStream_t stream)
{
    const float* x    = (const float*)d_in[0];
    const int*   ei   = (const int*)d_in[1];
    const float* ea   = (const float*)d_in[2];
    const float* resW = (const float*)d_in[3];
    const float* resB = (const float*)d_in[4];
    const float* conv[3][9];
    for (int l = 0; l < 3; l++)
        for (int p = 0; p < 9; p++)
            conv[l][p] = (const float*)d_in[5 + l * 9 + p];
    const float* gnp[3][3];
    for (int l = 0; l < 3; l++)
        for (int p = 0; p < 3; p++)
            gnp[l][p] = (const float*)d_in[32 + l * 3 + p];
    const float* gcnW = (const float*)d_in[41];
    const float* gcnB = (const float*)d_in[42];

    const int Nn = NN, E = NE;
    const int* src = ei;
    const int* dst = ei + E;

    float* ws = (float*)d_ws;
    float* XR   = ws; ws += (size_t)Nn * 512;
    float* XA   = ws; ws += (size_t)Nn * 512;
    float* XB   = ws; ws += (size_t)Nn * 512;
    float* QB   = ws; ws += (size_t)Nn * 512;
    float* KB   = ws; ws += (size_t)Nn * 512;
    float* VB   = ws; ws += (size_t)Nn * 512;
    float* ACC  = ws; ws += (size_t)Nn * 512;
    float* ALPH = ws; ws += (size_t)E * NHEAD;
    float* DEN  = ws; ws += (size_t)Nn * NHEAD;
    int*   AMAX = (int*)ws; ws += (size_t)Nn * NHEAD;
    float* COLS = ws; ws += 512;
    float* COLQ = ws; ws += 512;
    float* MU   = ws; ws += 512;
    float* ISTD = ws; ws += 512;
    float* DEG  = ws; ws += Nn;
    float* DINV = ws; ws += Nn;
    float* G    = ws; ws += 512 * 512;
    int*   MM   = (int*)ws; ws += 2;

    const int mtiles32 = Nn / 32;  // 625

    // x_res = x @ res_W + res_b
    gemm_wmma<<<(mtiles32 * (512 / 64) + 7) / 8, 256, 0, stream>>>(x, resW, resB, XR, Nn, 512, 512);

    const float* in = x;
    const int dins[3]  = {512, 256, 256};
    const int douts[3] = {256, 256, 512};
    const int dhs[3]   = {64, 64, 128};

    for (int l = 0; l < 3; l++) {
        int din = dins[l], dk = douts[l], dh = dhs[l];
        int gb = (mtiles32 * (dk / 64) + 7) / 8;
        gemm_wmma<<<gb, 256, 0, stream>>>(in, conv[l][0], conv[l][1], QB, Nn, din, dk);
        gemm_wmma<<<gb, 256, 0, stream>>>(in, conv[l][2], conv[l][3], KB, Nn, din, dk);
        gemm_wmma<<<gb, 256, 0, stream>>>(in, conv[l][4], conv[l][5], VB, Nn, din, dk);
        gemm_wmma<<<gb, 256, 0, stream>>>(in, conv[l][7], conv[l][8], XB, Nn, din, dk);

        fill_i32<<<(Nn * NHEAD + 255) / 256, 256, 0, stream>>>(AMAX, Nn * NHEAD, (int)0x80000000);
        fill_f32<<<(Nn * NHEAD + 255) / 256, 256, 0, stream>>>(DEN, Nn * NHEAD, 0.f);
        fill_f32<<<(Nn * dk + 255) / 256, 256, 0, stream>>>(ACC, Nn * dk, 0.f);

        int et = E * NHEAD;
        edge_alpha<<<(et + 255) / 256, 256, 0, stream>>>(QB, KB, conv[l][6], ea, src, dst, ALPH, AMAX, E, dk, dh);
        edge_accum<<<(et + 255) / 256, 256, 0, stream>>>(VB, conv[l][6], ea, src, dst, ALPH, AMAX, DEN, ACC, E, dk, dh);
        attn_finalize<<<(Nn * dk + 255) / 256, 256, 0, stream>>>(XB, ACC, DEN, l == 2 ? XR : nullptr, Nn, dk, dh);

        fill_f32<<<(dk + 255) / 256, 256, 0, stream>>>(COLS, dk, 0.f);
        fill_f32<<<(dk + 255) / 256, 256, 0, stream>>>(COLQ, dk, 0.f);
        dim3 sg((dk + 255) / 256, 100);
        gn_stats<<<sg, 256, 0, stream>>>(XB, COLS, COLQ, Nn, dk, 200);
        gn_final<<<(dk + 255) / 256, 256, 0, stream>>>(COLS, COLQ, gnp[l][2], MU, ISTD, Nn, dk);
        gn_apply<<<(Nn * dk + 255) / 256, 256, 0, stream>>>(XB, XA, MU, ISTD, gnp[l][2], gnp[l][0], gnp[l][1], Nn, dk);
        in = XA;
    }

    // GCN: deg, dinv, h = x@Wg, xt = self + scatter
    fill_f32<<<(Nn + 255) / 256, 256, 0, stream>>>(DEG, Nn, 2.f);
    gcn_deg<<<(E + 255) / 256, 256, 0, stream>>>(ea, dst, DEG, E);
    gcn_dinv<<<(Nn + 255) / 256, 256, 0, stream>>>(DEG, DINV, Nn);
    gemm_wmma<<<(mtiles32 * 8 + 7) / 8, 256, 0, stream>>>(XA, gcnW, nullptr, QB, Nn, 512, 512);
    gcn_self<<<(Nn * 512 + 255) / 256, 256, 0, stream>>>(QB, DINV, gcnB, KB, Nn, 512);
    gcn_scatter<<<(E * 32 + 255) / 256, 256, 0, stream>>>(QB, DINV, ea, src, dst, KB, E, 512);

    // G = xt^T @ xt (split-K=10, 32x32 tiles), min/max, triu
    fill_f32<<<(512 * 512 + 255) / 256, 256, 0, stream>>>(G, 512 * 512, 0.f);
    dim3 sk((16 * 16) / 8, 10);
    syrk_wmma<<<sk, 256, 0, stream>>>(KB, G, 512, Nn / 10);
    minmax_init<<<1, 1, 0, stream>>>(MM);
    minmax_red<<<256, 256, 0, stream>>>(G, 512 * 512, MM);
    triu_out<<<(512 * 512 + 255) / 256, 256, 0, stream>>>(G, MM, (float*)d_out, 512);
}